// Monarch_69587060130080
// MI455X (gfx1250) — compile-verified
//
#include <hip/hip_runtime.h>

typedef __attribute__((ext_vector_type(2))) float v2f;
typedef __attribute__((ext_vector_type(8))) float v8f;

#define COLS   8192
#define NROWS  4000
#define BB     64      // block size B
#define MM     63      // number of blocks M
#define PADCOL 32      // first zeroed input col of last R block (B - PAD)

// LDS layout for Y[k][j][c] (c = 16 cols per stripe), padded for bank conflicts:
// j-stride 20 floats  -> lane halves (j vs j+8) land 32 banks apart on stores
// k-stride 1296 floats-> rows 2 apart land 32 banks apart on phase-2 reads
#define JST 20
#define KST (BB * JST + 16)      // 1296 floats per k-plane
#define YFLOATS (MM * KST)       // 81648 floats = 326,592 B  (<= 320 KB LDS/WGP)

// ---------------- prep: R -> simplex rows, with post-padding zeroed ----------
__global__ __launch_bounds__(256) void prep_R(const float* __restrict__ Rp,
                                              float* __restrict__ Rn) {
    int t = blockIdx.x * blockDim.x + threadIdx.x;   // row id = k*64 + j
    if (t >= MM * BB) return;
    int k = t / BB;
    const float* src = Rp + (size_t)t * BB;
    float sum = 0.f;
    #pragma unroll 4
    for (int i = 0; i < BB; ++i) {
        float v = src[i];
        if (k == MM - 1 && i >= PADCOL) v = 0.f;
        sum += v * v;
    }
    float inv = 1.0f / sum;
    float* dst = Rn + (size_t)t * BB;
    #pragma unroll 4
    for (int i = 0; i < BB; ++i) {
        float v = src[i];
        if (k == MM - 1 && i >= PADCOL) v = 0.f;
        dst[i] = v * v * inv;                        // (p/||p||)^2 == p^2/sum(p^2)
    }
}

// -------- prep: L -> simplex rows, padded (64,64,64), zero row/col 63 --------
__global__ __launch_bounds__(256) void prep_L(const float* __restrict__ Lparams,
                                              float* __restrict__ Lp) {
    int t = blockIdx.x * blockDim.x + threadIdx.x;   // row id = j*64 + l
    if (t >= BB * BB) return;
    int j = t / BB, l = t % BB;
    float* dst = Lp + (size_t)t * BB;
    if (l >= MM) {                                   // padded l row -> zeros
        #pragma unroll 4
        for (int kk = 0; kk < BB; ++kk) dst[kk] = 0.f;
        return;
    }
    const float* src = Lparams + ((size_t)j * MM + l) * MM;
    float sum = 0.f;
    for (int kk = 0; kk < MM; ++kk) { float v = src[kk]; sum += v * v; }
    float inv = 1.0f / sum;
    for (int kk = 0; kk < MM; ++kk) dst[kk] = src[kk] * src[kk] * inv;
    dst[MM] = 0.f;                                   // padded k column -> zero
}

// ------------------------------- fused Monarch -------------------------------
// grid.x = COLS/16 (one 16-col stripe per workgroup), 256 threads = 8 wave32
__global__ __launch_bounds__(256) void monarch_main(const float* __restrict__ x,
                                                    const float* __restrict__ Rn,
                                                    const float* __restrict__ Lp,
                                                    float* __restrict__ out) {
    __shared__ float Ys[YFLOATS];
    const int ct   = blockIdx.x * 16;        // column stripe base
    const int wave = threadIdx.x >> 5;       // wave32
    const int lane = threadIdx.x & 31;
    const int lh   = lane >> 4;              // lane half (K>=2 half)
    const int lm   = lane & 15;              // m / n / c within tile

    // ---------------- phase 1: Y[k] (64x16) = R[k] (64x64) @ X[k] (64x16) ----
    for (int k = wave; k < MM; k += 8) {
        // Preload X block straight into WMMA-B layout:
        // reg r = 2*s+o holds X[k*64 + s*4 + 2*lh + o][ct+lm]  (zero past row 3999)
        float xr[32];
        #pragma unroll
        for (int r = 0; r < 32; ++r) {
            int s = r >> 1, o = r & 1;
            int grow = k * BB + s * 4 + 2 * lh + o;
            xr[r] = (grow < NROWS) ? x[(size_t)grow * COLS + ct + lm] : 0.f;
        }
        v8f acc[4] = {};
        const float* Rk = Rn + (size_t)k * BB * BB;
        #pragma unroll
        for (int s = 0; s < 16; ++s) {
            v2f b = { xr[2 * s], xr[2 * s + 1] };
            #pragma unroll
            for (int jt = 0; jt < 4; ++jt) {
                // A tile: lane lm = row m, float2 = K pair (2*lh, 2*lh+1), 8B aligned
                v2f a = *(const v2f*)(Rk + (size_t)(jt * 16 + lm) * BB + s * 4 + 2 * lh);
                acc[jt] = __builtin_amdgcn_wmma_f32_16x16x4_f32(
                    false, a, false, b, (short)0, acc[jt], false, false);
            }
        }
        #pragma unroll
        for (int jt = 0; jt < 4; ++jt) {
            #pragma unroll
            for (int v = 0; v < 8; ++v) {
                int j = jt * 16 + v + 8 * lh;        // D layout: M = v + 8*lh
                Ys[k * KST + j * JST + lm] = acc[jt][v];
            }
        }
    }
    __syncthreads();

    // ------------- phase 2: Z_j (64x16) = Lp[j] (64x64) @ Y_j (64x16) --------
    for (int j = wave; j < BB; j += 8) {
        v8f acc[4] = {};
        const float* Lj = Lp + (size_t)j * BB * BB;
        #pragma unroll
        for (int s = 0; s < 16; ++s) {
            int k0 = s * 4 + 2 * lh;
            // k=63 is padding: clamp LDS index; Lp[..][63]==0 kills the product
            int ka = (k0     > MM - 1) ? (MM - 1) : k0;
            int kb = (k0 + 1 > MM - 1) ? (MM - 1) : (k0 + 1);
            v2f b = { Ys[ka * KST + j * JST + lm],
                      Ys[kb * KST + j * JST + lm] };
            #pragma unroll
            for (int lt = 0; lt < 4; ++lt) {
                v2f a = *(const v2f*)(Lj + (size_t)(lt * 16 + lm) * BB + k0);
                acc[lt] = __builtin_amdgcn_wmma_f32_16x16x4_f32(
                    false, a, false, b, (short)0, acc[lt], false, false);
            }
        }
        #pragma unroll
        for (int lt = 0; lt < 4; ++lt) {
            #pragma unroll
            for (int v = 0; v < 8; ++v) {
                int l   = lt * 16 + v + 8 * lh;
                int row = l * BB + j;                // Z.reshape: row = l*64 + j
                if (row < NROWS)
                    out[(size_t)row * COLS + ct + lm] = acc[lt][v];
            }
        }
    }
}

extern "C" void kernel_launch(void* const* d_in, const int* in_sizes, int n_in,
                              void* d_out, int out_size, void* d_ws, size_t ws_size,
                              hipStream_t stream) {
    const float* x       = (const float*)d_in[0];   // (4000, 8192)
    const float* Lparams = (const float*)d_in[1];   // (64, 63, 63)
    const float* Rparams = (const float*)d_in[2];   // (63, 64, 64)
    float* out = (float*)d_out;                     // (4000, 8192)

    float* Rn = (float*)d_ws;                       // (63,64,64)  = 1,032,192 B
    float* Lp = Rn + (size_t)MM * BB * BB;          // (64,64,64)  = 1,048,576 B

    prep_R<<<(MM * BB + 255) / 256, 256, 0, stream>>>(Rparams, Rn);
    prep_L<<<(BB * BB + 255) / 256, 256, 0, stream>>>(Lparams, Lp);
    monarch_main<<<COLS / 16, 256, 0, stream>>>(x, Rn, Lp, out);
}